// MoEGate_86191403696185
// MI455X (gfx1250) — compile-verified
//
#include <hip/hip_runtime.h>
#include <hip/hip_bf16.h>
#include <math.h>
#include <stdint.h>

typedef float v2f __attribute__((ext_vector_type(2)));
typedef float v8f __attribute__((ext_vector_type(8)));

#define HDIM   2048
#define NEXP   64
#define SEQ    4096
#define NTOK   16384            // B*S
#define NBATCH 4

#define KC        64            // K-chunk in floats
#define NCHUNK    (HDIM / KC)   // 32
#define RSTRIDE   68            // padded LDS row stride (floats): 272B = 17*16 -> 16B aligned,
                                // and 2-bank-pair skew -> conflict-free ds_load_b64 fragments
#define A_FLOATS  (128 * RSTRIDE)   // 8704 floats
#define B_FLOATS  (64 * RSTRIDE)    // 4352 floats
#define BUF_FLOATS (A_FLOATS + B_FLOATS)

// ---- CDNA5 async global->LDS copy (ASYNCcnt-tracked DMA, 16B per lane) ----
__device__ __forceinline__ void async_cp16(const float* g, float* l)
{
    uint32_t lds = (uint32_t)(uintptr_t)(void*)l;   // flat shared ptr: low 32 bits = LDS offset
    asm volatile("global_load_async_to_lds_b128 %0, %1, off"
                 :: "v"(lds), "v"(g) : "memory");
}

template <int N>
__device__ __forceinline__ void wait_async()
{
    asm volatile("s_wait_asynccnt %0" :: "i"(N) : "memory");
}

// ---------------------------------------------------------------------------
// Kernel 1: fused gate.
// Block = 256 threads = 8 waves = 128 tokens (always inside one batch).
// Each wave: 16 tokens x 64 experts, f32 WMMA, operands staged in LDS via
// double-buffered async DMA. Weight tile shared by all 8 waves.
// ---------------------------------------------------------------------------
__global__ __launch_bounds__(256) void moe_gate_kernel(
    const float* __restrict__ hs,      // [NTOK, HDIM]
    const float* __restrict__ wgt,     // [NEXP, HDIM]
    float* __restrict__ out,           // [0,32768) idx-as-float, [32768,65536) weights
    int*   __restrict__ cntG,          // [NBATCH*NEXP]
    float* __restrict__ sumG)          // [NBATCH*NEXP]
{
    __shared__ float smem[2][BUF_FLOATS];   // [buf][ A(128x68) | B(64x68) ]  ~104KB
    __shared__ float lds_inv[128];
    __shared__ int   lds_cnt[NEXP];

    const int tid  = threadIdx.x;
    const int wid  = tid >> 5;
    const int lane = tid & 31;
    const int m    = lane & 15;     // row-in-tile (A) / expert-in-tile (B)
    const int h    = lane >> 4;     // K-pair selector per ISA fragment layout
    const int blockTok = blockIdx.x * 128;

    if (tid < NEXP) lds_cnt[tid] = 0;

    const float* gA = hs + (size_t)blockTok * HDIM;

    // Issue one K-chunk of A (128 rows) + B (64 rows) as async B128 copies.
    // 12 instructions per thread per chunk; fully coalesced (16 threads = 256B row).
    auto issue_chunk = [&](int c, int buf) {
        float* lA = smem[buf];
        float* lB = smem[buf] + A_FLOATS;
        const int seg = tid & 15;          // 16B segment within the 256B row-chunk
        const int r0  = tid >> 4;          // 0..15
        const float* gac = gA  + c * KC + seg * 4;
        const float* gbc = wgt + c * KC + seg * 4;
#pragma unroll
        for (int it = 0; it < 8; ++it) {   // A: 128 rows
            int row = r0 + it * 16;
            async_cp16(gac + (size_t)row * HDIM, lA + row * RSTRIDE + seg * 4);
        }
#pragma unroll
        for (int it = 0; it < 4; ++it) {   // B: 64 rows
            int row = r0 + it * 16;
            async_cp16(gbc + (size_t)row * HDIM, lB + row * RSTRIDE + seg * 4);
        }
    };

    v8f c[4];
#pragma unroll
    for (int et = 0; et < 4; ++et) c[et] = (v8f)0.0f;

    issue_chunk(0, 0);
    issue_chunk(1, 1);

    const int arow = wid * 16 + m;         // this lane's A row (block-local token)
    for (int ch = 0; ch < NCHUNK; ++ch) {
        const int buf = ch & 1;
        if (ch == NCHUNK - 1) wait_async<0>();   // only last chunk outstanding
        else                  wait_async<12>();  // chunk ch done, ch+1 still in flight
        __syncthreads();

        const float* lA = smem[buf] + arow * RSTRIDE + 2 * h;
        const float* lB = smem[buf] + A_FLOATS + m * RSTRIDE + 2 * h;
#pragma unroll 4
        for (int j = 0; j < KC / 4; ++j) {
            v2f a = *(const v2f*)(lA + j * 4);
#pragma unroll
            for (int et = 0; et < 4; ++et) {
                v2f b = *(const v2f*)(lB + et * 16 * RSTRIDE + j * 4);
                c[et] = __builtin_amdgcn_wmma_f32_16x16x4_f32(
                    false, a, false, b, (short)0, c[et], false, false);
            }
        }
        __syncthreads();
        if (ch + 2 < NCHUNK) issue_chunk(ch + 2, buf);
    }

    // Spill C tiles into (now free) buffer 0: logits[128][RSTRIDE].
    // VGPR r: lanes0-15 -> M=r, lanes16-31 -> M=8+r.
    float* logits = smem[0];
#pragma unroll
    for (int et = 0; et < 4; ++et)
#pragma unroll
        for (int r = 0; r < 8; ++r) {
            int row = wid * 16 + (h ? (8 + r) : r);
            logits[row * RSTRIDE + et * 16 + m] = c[et][r];
        }
    __syncthreads();

    // One thread per token: softmax + top-2 (JAX tie-break: lowest index wins).
    if (tid < 128) {
        float* row = logits + tid * RSTRIDE;
        float mx = -INFINITY;
#pragma unroll 8
        for (int e = 0; e < NEXP; ++e) mx = fmaxf(mx, row[e]);

        float sum = 0.0f;
        float v1 = -INFINITY, v2 = -INFINITY;
        int   i1 = 0, i2 = 0;
        for (int e = 0; e < NEXP; ++e) {
            float v = expf(row[e] - mx);
            row[e] = v;                 // keep numerators for mean-score pass
            sum += v;
            if (v > v1)      { v2 = v1; i2 = i1; v1 = v; i1 = e; }
            else if (v > v2) { v2 = v;  i2 = e; }
        }
        const float inv = 1.0f / sum;
        lds_inv[tid] = inv;

        const float p1 = v1 * inv, p2 = v2 * inv;
        const float dn = 1.0f / (p1 + p2 + 1e-20f);

        const int t = blockTok + tid;
        out[t * 2 + 0] = (float)i1;
        out[t * 2 + 1] = (float)i2;
        out[2 * NTOK + t * 2 + 0] = p1 * dn;
        out[2 * NTOK + t * 2 + 1] = p2 * dn;

        atomicAdd(&lds_cnt[i1], 1);
        atomicAdd(&lds_cnt[i2], 1);
    }
    __syncthreads();

    // Per-block expert partials; one global atomic per (expert, stat).
    const int b = blockTok / SEQ;   // whole block lies in one batch (128 | 4096)
    if (tid < NEXP) {
        float acc = 0.0f;
        for (int tt = 0; tt < 128; ++tt)
            acc += logits[tt * RSTRIDE + tid] * lds_inv[tt];
        atomicAdd(&sumG[b * NEXP + tid], acc);
        atomicAdd(&cntG[b * NEXP + tid], lds_cnt[tid]);
    }
}

// ---------------------------------------------------------------------------
// Kernel 2: fold stats into the scalar aux loss.
// aux = ALPHA * mean_b( sum_e  (cnt/128) * (sums/S) )
// ---------------------------------------------------------------------------
__global__ void aux_loss_kernel(const int* __restrict__ cnt,
                                const float* __restrict__ sums,
                                float* __restrict__ out_aux)
{
    __shared__ float red[256];
    const int i = threadIdx.x;           // (b,e) flat, 256 = 4*64
    red[i] = (float)cnt[i] * (1.0f / 128.0f) * (sums[i] * (1.0f / (float)SEQ));
    __syncthreads();
    for (int s = 128; s > 0; s >>= 1) {
        if (i < s) red[i] += red[i + s];
        __syncthreads();
    }
    if (i == 0) out_aux[0] = red[0] * (0.1f / (float)NBATCH);
}

__global__ void zero_ws_kernel(int* cnt, float* sums)
{
    int i = threadIdx.x;   // 256 threads
    cnt[i]  = 0;
    sums[i] = 0.0f;
}

extern "C" void kernel_launch(void* const* d_in, const int* in_sizes, int n_in,
                              void* d_out, int out_size, void* d_ws, size_t ws_size,
                              hipStream_t stream)
{
    const float* hs  = (const float*)d_in[0];   // [4,4096,2048] f32
    const float* wgt = (const float*)d_in[1];   // [64,2048] f32
    // d_in[2] = top_k (always 2 for this reference)
    float* out = (float*)d_out;                 // 65537 floats

    int*   cntG = (int*)d_ws;
    float* sumG = (float*)((char*)d_ws + NBATCH * NEXP * sizeof(int));

    zero_ws_kernel<<<1, NBATCH * NEXP, 0, stream>>>(cntG, sumG);
    moe_gate_kernel<<<NTOK / 128, 256, 0, stream>>>(hs, wgt, out, cntG, sumG);
    aux_loss_kernel<<<1, NBATCH * NEXP, 0, stream>>>(cntG, sumG, out + 4 * NTOK);
}